// GPTJAttention_5162550690215
// MI455X (gfx1250) — compile-verified
//
#include <hip/hip_runtime.h>
#include <math.h>

// ---------------- problem constants ----------------
#define BB   2
#define SS   2048
#define HH   4096
#define NH   16
#define DH   256
#define ROT  64
#define N3   12288          // 3*H
#define MTOT 4096           // B*S
#define PFD  16             // prefetch distance in K-steps (x4 elements)

typedef __attribute__((ext_vector_type(2))) float v2f;
typedef __attribute__((ext_vector_type(8))) float v8f;

// D = A(16x4) * B(4x16) + C, fp32 WMMA.
// A frag: a[i] = A[lane%16][2*(lane/16)+i]
// B frag: b[i] = B[2*(lane/16)+i][lane%16]
// C/D   : c[v] = C[v + 8*(lane/16)][lane%16]
__device__ __forceinline__ v8f wmma_f32(v2f a, v2f b, v8f c) {
    return __builtin_amdgcn_wmma_f32_16x16x4_f32(false, a, false, b, (short)0, c,
                                                 false, false);
}

// =====================================================================
// Generic 64x64-per-wave fp32 WMMA GEMM body.
// block = 128 threads (2x2 waves); block tile = 128(M) x 128(N).
// Each wave: 4x4 grid of 16x16 C tiles (128 acc VGPRs).
// =====================================================================
template <int NDIM>
__device__ __forceinline__ void gemm_body(const float* __restrict__ X,
                                          const float* __restrict__ W,
                                          int m0, int n0, int l16, int half,
                                          v8f (&acc)[4][4])
{
    const float* __restrict__ Arow0 = X + (size_t)(m0 + l16) * HH;

    for (int k0 = 0; k0 < HH; k0 += 4) {
        const int ka = k0 + 2 * half;

        // per-lane prefetch of the B rows PFD k-steps ahead (global_prefetch_b8)
        const float* __restrict__ Bpf =
            W + (size_t)(ka + 4 * PFD) * NDIM + n0 + l16;
        __builtin_prefetch(Bpf, 0, 1);

        v2f a[4];
#pragma unroll
        for (int tm = 0; tm < 4; ++tm) {
            const float* __restrict__ Ap = Arow0 + (size_t)tm * 16 * HH + ka;
            a[tm].x = Ap[0]; a[tm].y = Ap[1];          // global_load_b64
        }
        v2f b[4];
        const float* __restrict__ Bp = W + (size_t)ka * NDIM + n0 + l16;
#pragma unroll
        for (int tn = 0; tn < 4; ++tn) {
            b[tn].x = Bp[tn * 16];
            b[tn].y = Bp[tn * 16 + NDIM];
        }
#pragma unroll
        for (int tm = 0; tm < 4; ++tm)
#pragma unroll
            for (int tn = 0; tn < 4; ++tn)
                acc[tm][tn] = wmma_f32(a[tm], b[tn], acc[tm][tn]);
    }
}

// =====================================================================
// Kernel 1: QKV GEMM.  X[MTOT,HH] x Wqkv[HH,N3] -> Q/K/V in [B,NH,S,DH].
// grid = (N3/128, MTOT/128)
// =====================================================================
__global__ __launch_bounds__(128)
void qkv_gemm_kernel(const float* __restrict__ X, const float* __restrict__ W,
                     float* __restrict__ Qo, float* __restrict__ Ko,
                     float* __restrict__ Vo)
{
    const int lane  = threadIdx.x & 31;
    const int wave  = threadIdx.x >> 5;
    const int l16   = lane & 15;
    const int half  = lane >> 4;
    const int waveM = wave >> 1, waveN = wave & 1;
    const int m0 = blockIdx.y * 128 + waveM * 64;
    const int n0 = blockIdx.x * 128 + waveN * 64;

    v8f acc[4][4] = {};
    gemm_body<N3>(X, W, m0, n0, l16, half, acc);

    // scatter epilogue into [B, NH, S, DH] per q/k/v
#pragma unroll
    for (int tm = 0; tm < 4; ++tm) {
#pragma unroll
        for (int tn = 0; tn < 4; ++tn) {
            const int n = n0 + tn * 16 + l16;
            const int which = n >> 12;          // 0=q 1=k 2=v
            const int rem   = n & 4095;
            const int h     = rem >> 8;
            const int d     = rem & 255;
            float* __restrict__ dst = (which == 0) ? Qo : (which == 1) ? Ko : Vo;
#pragma unroll
            for (int v = 0; v < 8; ++v) {
                const int m = m0 + tm * 16 + v + 8 * half;
                const int b = m >> 11, s = m & 2047;
                dst[(((size_t)b * NH + h) * SS + s) * DH + d] = acc[tm][tn][v];
            }
        }
    }
}

// =====================================================================
// Kernel 2: GPT-J interleaved RoPE on first ROT dims of Q and K.
// one thread per (b,h,s,pair).  total = B*NH*S*(ROT/2)
// =====================================================================
__global__ void rope_kernel(float* __restrict__ Q, float* __restrict__ K,
                            const int* __restrict__ pos)
{
    const int idx = blockIdx.x * blockDim.x + threadIdx.x;
    const int i = idx & 31;                 // pair index 0..31
    const int s = (idx >> 5) & 2047;
    const int h = (idx >> 16) & 15;
    const int b = idx >> 20;

    const float inv_freq = __powf(10000.0f, -(float)(2 * i) / (float)ROT);
    const float f = (float)pos[b * SS + s] * inv_freq;
    float sn, cs;
    __sincosf(f, &sn, &cs);

    const size_t base = (((size_t)b * NH + h) * SS + s) * DH + 2 * i;
    const float q1 = Q[base], q2 = Q[base + 1];
    Q[base]     = q1 * cs - q2 * sn;
    Q[base + 1] = q2 * cs + q1 * sn;
    const float k1 = K[base], k2 = K[base + 1];
    K[base]     = k1 * cs - k2 * sn;
    K[base + 1] = k2 * cs + k1 * sn;
}

// =====================================================================
// Kernel 3: causal flash attention, fp32 WMMA.
// block = 128 (4 waves); wave handles 16 query rows, full DH=256 in regs.
// grid = (S/64, NH, B)
// =====================================================================
__global__ __launch_bounds__(128)
void flash_attn_kernel(const float* __restrict__ Q, const float* __restrict__ K,
                       const float* __restrict__ V, float* __restrict__ O)
{
    const int lane = threadIdx.x & 31;
    const int wave = threadIdx.x >> 5;
    const int l16  = lane & 15;
    const int half = lane >> 4;
    const int h  = blockIdx.y;
    const int b  = blockIdx.z;
    const int q0 = blockIdx.x * 64 + wave * 16;

    const float scale = 0.0625f;            // DH^-0.5 = 1/16
    const float* __restrict__ Qh = Q + (((size_t)b * NH + h) * SS) * DH;
    const float* __restrict__ Kh = K + (((size_t)b * NH + h) * SS) * DH;
    const float* __restrict__ Vh = V + (((size_t)b * NH + h) * SS) * DH;

    __shared__ float lds_p[4][16][17];      // per-wave P staging (C->A transpose)

    v8f acc[16] = {};                       // 16 rows x 256 cols output
    float mrow[8], lrow[8];
#pragma unroll
    for (int v = 0; v < 8; ++v) { mrow[v] = -1.0e30f; lrow[v] = 0.0f; }

    const float* __restrict__ Qrow = Qh + (size_t)(q0 + l16) * DH;

    for (int kk = 0; kk <= q0; kk += 16) {
        // prefetch next chunk's K and V rows while we compute this one
        __builtin_prefetch(Kh + (size_t)(kk + 16 + l16) * DH + 2 * half, 0, 1);
        __builtin_prefetch(Vh + (size_t)(kk + 16 + l16) * DH + 2 * half, 0, 1);

        // ---- S = Q * K^T  (16x16 tile) ----
        v8f sc = {};
#pragma unroll 8
        for (int d0 = 0; d0 < DH; d0 += 4) {
            const int da = d0 + 2 * half;
            v2f a; a.x = Qrow[da]; a.y = Qrow[da + 1];
            const float* __restrict__ Kp = Kh + (size_t)(kk + l16) * DH + da;
            v2f bf; bf.x = Kp[0]; bf.y = Kp[1];
            sc = wmma_f32(a, bf, sc);
        }

        // ---- scale + causal mask + online softmax ----
        const int ki = kk + l16;            // key col this lane holds
        float corr[8];
#pragma unroll
        for (int v = 0; v < 8; ++v) {
            const int qi = q0 + v + 8 * half;
            float sv = sc[v] * scale;
            if (ki > qi) sv = -1.0e30f;

            float x = sv;
            x = fmaxf(x, __shfl_xor(x, 1, 16));
            x = fmaxf(x, __shfl_xor(x, 2, 16));
            x = fmaxf(x, __shfl_xor(x, 4, 16));
            x = fmaxf(x, __shfl_xor(x, 8, 16));
            const float mnew = fmaxf(mrow[v], x);
            corr[v] = __expf(mrow[v] - mnew);
            mrow[v] = mnew;

            const float p = __expf(sv - mnew);
            float ps = p;
            ps += __shfl_xor(ps, 1, 16);
            ps += __shfl_xor(ps, 2, 16);
            ps += __shfl_xor(ps, 4, 16);
            ps += __shfl_xor(ps, 8, 16);
            lrow[v] = lrow[v] * corr[v] + ps;

            lds_p[wave][v + 8 * half][l16] = p;   // stage P in row-major
        }
#pragma unroll
        for (int t = 0; t < 16; ++t)
#pragma unroll
            for (int v = 0; v < 8; ++v) acc[t][v] *= corr[v];

        // ---- O += P * V  (16x16 P, 16x256 V tile) ----
#pragma unroll
        for (int k0 = 0; k0 < 16; k0 += 4) {
            const int kp = k0 + 2 * half;
            v2f a; a.x = lds_p[wave][l16][kp]; a.y = lds_p[wave][l16][kp + 1];
            const float* __restrict__ Vp = Vh + (size_t)(kk + kp) * DH + l16;
#pragma unroll
            for (int t = 0; t < 16; ++t) {
                v2f bf; bf.x = Vp[t * 16]; bf.y = Vp[t * 16 + DH];
                acc[t] = wmma_f32(a, bf, acc[t]);
            }
        }
    }

    // ---- normalize and write [B,S,NH,DH] ----
    float inv[8];
#pragma unroll
    for (int v = 0; v < 8; ++v) inv[v] = 1.0f / lrow[v];
#pragma unroll
    for (int t = 0; t < 16; ++t) {
        const int d = t * 16 + l16;
#pragma unroll
        for (int v = 0; v < 8; ++v) {
            const int qi = q0 + v + 8 * half;
            O[(((size_t)b * SS + qi) * NH + h) * DH + d] = acc[t][v] * inv[v];
        }
    }
}

// =====================================================================
// Kernel 4: output GEMM.  A[MTOT,HH] x Wout[HH,HH] -> out[MTOT,HH]
// grid = (HH/128, MTOT/128)
// =====================================================================
__global__ __launch_bounds__(128)
void out_gemm_kernel(const float* __restrict__ A_, const float* __restrict__ W,
                     float* __restrict__ Out)
{
    const int lane  = threadIdx.x & 31;
    const int wave  = threadIdx.x >> 5;
    const int l16   = lane & 15;
    const int half  = lane >> 4;
    const int waveM = wave >> 1, waveN = wave & 1;
    const int m0 = blockIdx.y * 128 + waveM * 64;
    const int n0 = blockIdx.x * 128 + waveN * 64;

    v8f acc[4][4] = {};
    gemm_body<HH>(A_, W, m0, n0, l16, half, acc);

#pragma unroll
    for (int tm = 0; tm < 4; ++tm) {
#pragma unroll
        for (int tn = 0; tn < 4; ++tn) {
            const int n = n0 + tn * 16 + l16;
#pragma unroll
            for (int v = 0; v < 8; ++v) {
                const int m = m0 + tm * 16 + v + 8 * half;
                Out[(size_t)m * HH + n] = acc[tm][tn][v];
            }
        }
    }
}

// =====================================================================
extern "C" void kernel_launch(void* const* d_in, const int* in_sizes, int n_in,
                              void* d_out, int out_size, void* d_ws, size_t ws_size,
                              hipStream_t stream)
{
    const float* hidden = (const float*)d_in[0];
    const int*   pos    = (const int*)d_in[1];
    const float* Wqkv   = (const float*)d_in[2];
    const float* Wout   = (const float*)d_in[3];
    float* out = (float*)d_out;

    const size_t perBuf = (size_t)BB * NH * SS * DH;   // 16,777,216 floats
    float* q    = (float*)d_ws;
    float* k    = q + perBuf;
    float* v    = k + perBuf;
    float* attn = v + perBuf;

    dim3 qkvGrid(N3 / 128, MTOT / 128);                // (96, 32)
    qkv_gemm_kernel<<<qkvGrid, 128, 0, stream>>>(hidden, Wqkv, q, k, v);

    const int ropeTot = BB * NH * SS * (ROT / 2);      // 2,097,152
    rope_kernel<<<ropeTot / 256, 256, 0, stream>>>(q, k, pos);

    dim3 aGrid(SS / 64, NH, BB);                       // (32, 16, 2)
    flash_attn_kernel<<<aGrid, 128, 0, stream>>>(q, k, v, attn);

    dim3 oGrid(HH / 128, MTOT / 128);                  // (32, 32)
    out_gemm_kernel<<<oGrid, 128, 0, stream>>>(attn, Wout, out);
}